// Nabla4d_44848048504983
// MI455X (gfx1250) — compile-verified
//
#include <hip/hip_runtime.h>

// nabla_4d on [T=32, Z=64, Y=128, X=128] f32 -> [4, T, Z, Y, X] f32.
// Bandwidth-bound (0.2 FLOP/byte): stream input once, write 4x output.
// gy is computed per 16x16 (y,x) tile as GY = Dy * X with
// V_WMMA_F32_16X16X4_F32 (4 chained K=4 chunks); Dy is the constant
// bidiagonal difference matrix built in registers. X sits in the WMMA
// B-operand layout, which loads perfectly coalesced from row-major memory.
// gx/gz/gt use shifted/neighbor loads (L2 hits; whole input fits in 192MB L2).

typedef __attribute__((ext_vector_type(2))) float v2f;
typedef __attribute__((ext_vector_type(8))) float v8f;

#define NT 32
#define NZ 64
#define NY 128
#define NX 128
#define NELEM (NT * NZ * NY * NX) // 33554432

__global__ __launch_bounds__(256) void nabla4d_wmma_kernel(
    const float* __restrict__ X, float* __restrict__ out) {
  const int lane = threadIdx.x & 31;
  const int waveInBlock = threadIdx.x >> 5;
  const int w = blockIdx.x * 8 + waveInBlock; // tile id, 0..131071

  // w = ((t*NZ + z)*8 + yt)*8 + xt
  const int xt = w & 7;
  const int yt = (w >> 3) & 7;
  const int tz = w >> 6;   // t*NZ + z, 0..2047
  const int z  = tz & 63;
  const int t  = tz >> 6;

  const int x0 = xt << 4;
  const int y0 = yt << 4;

  const int n    = lane & 15;  // column within tile (WMMA N index)
  const int half = lane >> 4;  // half-wave select
  const int m    = n;          // A-operand row index carried by this lane

  const unsigned slab  = (unsigned)tz * (NY * NX);
  const unsigned zslab = (unsigned)(t * NZ + (z < NZ - 1 ? z + 1 : z)) * (NY * NX);
  const unsigned tslab = (unsigned)((t < NT - 1 ? t + 1 : t) * NZ + z) * (NY * NX);

  const int col  = x0 + n;
  const int colp = (col < NX - 1) ? col + 1 : col; // clamped x+1 (value masked below)

  float b[8];   // X tile, WMMA B-operand layout (row 4c+2*half+j per reg)
  float bx[8];  // X shifted +1 in x
  float bz[8];  // X at z+1 (clamped)
  float bt[8];  // X at t+1 (clamped)
  float a[8];   // Dy, WMMA A-operand layout (K = 4c+2*half+j per reg)
  int   rows[8];
#pragma unroll
  for (int k = 0; k < 8; ++k) {
    const int c = k >> 1, j = k & 1;
    const int kk = 4 * c + 2 * half + j; // K index == tile row index
    rows[k] = kk;
    const unsigned rowoff = (unsigned)(y0 + kk) * NX;
    b[k]  = X[slab  + rowoff + col];
    bx[k] = X[slab  + rowoff + colp];
    bz[k] = X[zslab + rowoff + col];
    bt[k] = X[tslab + rowoff + col];
    // Dy[m][k]: -1 on diagonal, +1 on superdiagonal
    a[k] = (kk == m) ? -1.0f : ((kk == m + 1) ? 1.0f : 0.0f);
  }

  // Row y0+16 (clamped) for the gy m=15 fixup.
  const int  yr16   = (y0 + 16 < NY) ? (y0 + 16) : (NY - 1);
  const bool lastY  = (y0 + 16 >= NY);
  const float xrow16 = X[slab + (unsigned)yr16 * NX + col];

  // GY = Dy * Xtile via 4 chained 16x16x4 f32 WMMAs (K = 16).
  // EXEC is all-1s here: no divergent control flow above, grid covers exactly.
  v8f acc = {};
#pragma unroll
  for (int c = 0; c < 4; ++c) {
    v2f av, bv;
    av.x = a[2 * c];  av.y = a[2 * c + 1];
    bv.x = b[2 * c];  bv.y = b[2 * c + 1];
    acc = __builtin_amdgcn_wmma_f32_16x16x4_f32(
        /*neg_a=*/false, av, /*neg_b=*/false, bv,
        /*c_mod=*/(short)0, acc, /*reuse_a=*/false, /*reuse_b=*/false);
  }

  // C/D layout: reg r holds gy[M = r + 8*half][n]. M=15 (r=7, half=1) needs
  // row 16 which is outside the K range -> patch with a select (no EXEC change).
  float d[8];
#pragma unroll
  for (int r = 0; r < 8; ++r) d[r] = acc[r];
  {
    // b[7] on half==1 is tile row 15 (k=4*3+2*1+1).
    const float fix = lastY ? 0.0f : (xrow16 - b[7]);
    d[7] = half ? fix : d[7];
  }

  float* __restrict__ gx = out;
  float* __restrict__ gy = out + (unsigned)NELEM;
  float* __restrict__ gz = out + 2u * (unsigned)NELEM;
  float* __restrict__ gt = out + 3u * (unsigned)NELEM;

  const bool okx = (col < NX - 1);
  const bool okz = (z < NZ - 1);
  const bool okt = (t < NT - 1);

#pragma unroll
  for (int k = 0; k < 8; ++k) {
    const unsigned off = slab + (unsigned)(y0 + rows[k]) * NX + col;
    gx[off] = okx ? (bx[k] - b[k]) : 0.0f;
    gz[off] = okz ? (bz[k] - b[k]) : 0.0f;
    gt[off] = okt ? (bt[k] - b[k]) : 0.0f;
  }
#pragma unroll
  for (int r = 0; r < 8; ++r) {
    const unsigned off = slab + (unsigned)(y0 + r + 8 * half) * NX + col;
    gy[off] = d[r];
  }
}

extern "C" void kernel_launch(void* const* d_in, const int* in_sizes, int n_in,
                              void* d_out, int out_size, void* d_ws, size_t ws_size,
                              hipStream_t stream) {
  const float* X = (const float*)d_in[0];
  float* out = (float*)d_out;
  // 131072 wave-tiles / 8 waves per 256-thread block = 16384 blocks.
  nabla4d_wmma_kernel<<<16384, 256, 0, stream>>>(X, out);
}